// VN_DGCNN_71468255805656
// MI455X (gfx1250) — compile-verified
//
#include <hip/hip_runtime.h>
#include <hip/hip_bf16.h>
#include <math.h>

#define BATCH 8
#define NPTS  4096
#define KNN   20
#define VCOUT 21
#define FD    1024
#define SLOPE 0.2f
#define VEPS  1e-6f
#define BNEPS 1e-5f

typedef float v2f __attribute__((ext_vector_type(2)));
typedef float v8f __attribute__((ext_vector_type(8)));

static __device__ __forceinline__ v8f wmma_f32_16x16x4(v2f a, v2f b, v8f c) {
  // D = A(16x4) * B(4x16) + C(16x16), fp32 WMMA (CDNA5)
  return __builtin_amdgcn_wmma_f32_16x16x4_f32(false, a, false, b, (short)0, c, false, false);
}

// ---------------------------------------------------------------------------
// xx[b][n] = sum_c src[b][c][n]^2   (src viewed as (B, C3, N) with batch stride)
// ---------------------------------------------------------------------------
__global__ __launch_bounds__(256) void sqnorm_kernel(
    const float* __restrict__ src, long bstride, int C3, float* __restrict__ xx)
{
  const long t = (long)blockIdx.x * 256 + threadIdx.x;   // t < B*N
  const int b = (int)(t / NPTS);
  const int n = (int)(t % NPTS);
  const float* p = src + (long)b * bstride + n;
  float s = 0.f;
  for (int c = 0; c < C3; ++c) {
    const float v = p[(long)c * NPTS];
    s += v * v;
  }
  xx[t] = s;
}

// ---------------------------------------------------------------------------
// Fused KNN: one workgroup computes a 16 x 4096 strip of neg_dist via
// V_WMMA_F32_16X16X4_F32, keeps it entirely in LDS (256KB of the 320KB WGP
// LDS), then does iterative top-20 argmax per row. Only indices hit memory.
// ---------------------------------------------------------------------------
template<int C3>
__global__ __launch_bounds__(256) void knn_topk_kernel(
    const float* __restrict__ src, long bstride,
    const float* __restrict__ xx, int* __restrict__ idxOut)
{
  constexpr int NCH = (C3 + 3) / 4;
  extern __shared__ float lds[];
  float* strip = lds;                 // [16][NPTS]
  float* sxr   = lds + 16 * NPTS;     // [16] row sq-norms
  const int b = blockIdx.y;
  const int rowBase = blockIdx.x * 16;
  const int tid = threadIdx.x;
  const int wave = tid >> 5;
  const int lane = tid & 31;
  const int laneHalf = lane & 15;
  const int hi = (lane >= 16) ? 1 : 0;
  const float* X = src + (long)b * bstride;
  const float* xxb = xx + b * NPTS;

  if (tid < 16) sxr[tid] = xxb[rowBase + tid];
  __syncthreads();

  // A fragments: rows rowBase..rowBase+15 of X^T (16 x C3), reused all tiles
  v2f afrag[NCH];
  {
    const int m = rowBase + laneHalf;
    #pragma unroll
    for (int c = 0; c < NCH; ++c) {
      const int k0 = c * 4 + hi * 2;
      afrag[c].x = (k0     < C3) ? X[(long)(k0    ) * NPTS + m] : 0.f;
      afrag[c].y = (k0 + 1 < C3) ? X[(long)(k0 + 1) * NPTS + m] : 0.f;
    }
  }

  for (int colBase = wave * 16; colBase < NPTS; colBase += 128) {
    const int n = colBase + laneHalf;
    v8f acc = {0.f,0.f,0.f,0.f,0.f,0.f,0.f,0.f};
    #pragma unroll
    for (int c = 0; c < NCH; ++c) {
      const int k0 = c * 4 + hi * 2;
      v2f bfrag;
      bfrag.x = (k0     < C3) ? X[(long)(k0    ) * NPTS + n] : 0.f;
      bfrag.y = (k0 + 1 < C3) ? X[(long)(k0 + 1) * NPTS + n] : 0.f;
      acc = wmma_f32_16x16x4(afrag[c], bfrag, acc);
    }
    const float xxc = xxb[n];
    #pragma unroll
    for (int g = 0; g < 8; ++g) {
      const int m = g + hi * 8;
      strip[m * NPTS + n] = 2.f * acc[g] - sxr[m] - xxc;   // neg squared dist
    }
  }
  __syncthreads();

  // top-20 per row: 8 waves x 2 rows, iterative argmax with -inf exclusion
  #pragma unroll
  for (int rr = 0; rr < 2; ++rr) {
    const int r = wave * 2 + rr;
    float* row = strip + r * NPTS;
    int* outp = idxOut + ((long)(b * NPTS + rowBase + r)) * KNN;
    for (int sel = 0; sel < KNN; ++sel) {
      float best = -3.4e38f;
      int bidx = NPTS;
      for (int j = lane; j < NPTS; j += 32) {
        const float v = row[j];
        if (v > best || (v == best && j < bidx)) { best = v; bidx = j; }
      }
      #pragma unroll
      for (int off = 16; off >= 1; off >>= 1) {
        const float ov = __shfl_xor(best, off, 32);
        const int   oi = __shfl_xor(bidx, off, 32);
        if (ov > best || (ov == best && oi < bidx)) { best = ov; bidx = oi; }
      }
      bidx = __shfl(bidx, 0, 32);
      if (lane == 0) { outp[sel] = bidx; row[bidx] = -3.4e38f; }
      __threadfence_block();
    }
  }
}

// ---------------------------------------------------------------------------
// VN layer pass 1: p = Wf * feat ; store p (B,21,3,N,K) and accumulate
// per-channel (sum ||p||, sum ||p||^2) via LDS ds_add_f32 then global atomic.
// GF=true: feat built on-the-fly from (B,CIN0,3,N) src + idx (diff | center).
// GF=false: direct input (B,CIN0,3,N,K).
// ---------------------------------------------------------------------------
template<int CIN0, bool GF>
__global__ __launch_bounds__(256) void vn_p_kernel(
    const float* __restrict__ src, long srcBStride,
    const int* __restrict__ idx,
    const float* __restrict__ W,
    float* __restrict__ pOut,
    float* __restrict__ stats)
{
  constexpr int CIN = GF ? 2 * CIN0 : CIN0;
  __shared__ float sW[VCOUT * CIN];
  __shared__ float sStat[VCOUT * 2];
  for (int i = threadIdx.x; i < VCOUT * CIN; i += 256) sW[i] = W[i];
  if (threadIdx.x < VCOUT * 2) sStat[threadIdx.x] = 0.f;
  __syncthreads();

  const long t = (long)blockIdx.x * 256 + threadIdx.x;   // t < B*N*K exactly
  const int b = (int)(t / ((long)NPTS * KNN));
  const int rem = (int)(t % ((long)NPTS * KNN));
  const int n = rem / KNN;
  const int k = rem % KNN;

  float acc[VCOUT][3];
  #pragma unroll
  for (int o = 0; o < VCOUT; ++o) { acc[o][0] = 0.f; acc[o][1] = 0.f; acc[o][2] = 0.f; }

  if constexpr (GF) {
    const int j = idx[t];
    const float* Xb = src + (long)b * srcBStride;
    for (int c = 0; c < CIN0; ++c) {               // neighbor - center channels
      const float* base = Xb + (long)(c * 3) * NPTS;
      const float f0 = base[j]            - base[n];
      const float f1 = base[NPTS + j]     - base[NPTS + n];
      const float f2 = base[2 * NPTS + j] - base[2 * NPTS + n];
      #pragma unroll
      for (int o = 0; o < VCOUT; ++o) {
        const float w = sW[o * CIN + c];
        acc[o][0] += w * f0; acc[o][1] += w * f1; acc[o][2] += w * f2;
      }
    }
    for (int c = 0; c < CIN0; ++c) {               // center channels
      const float* base = Xb + (long)(c * 3) * NPTS;
      const float f0 = base[n], f1 = base[NPTS + n], f2 = base[2 * NPTS + n];
      #pragma unroll
      for (int o = 0; o < VCOUT; ++o) {
        const float w = sW[o * CIN + CIN0 + c];
        acc[o][0] += w * f0; acc[o][1] += w * f1; acc[o][2] += w * f2;
      }
    }
  } else {
    for (int c = 0; c < CIN0; ++c) {
      const long base = (((long)(b * CIN0 + c) * 3) * NPTS + n) * KNN + k;
      const float f0 = src[base];
      const float f1 = src[base + (long)NPTS * KNN];
      const float f2 = src[base + 2L * NPTS * KNN];
      #pragma unroll
      for (int o = 0; o < VCOUT; ++o) {
        const float w = sW[o * CIN + c];
        acc[o][0] += w * f0; acc[o][1] += w * f1; acc[o][2] += w * f2;
      }
    }
  }

  #pragma unroll
  for (int o = 0; o < VCOUT; ++o) {
    const long base = (((long)(b * VCOUT + o) * 3) * NPTS + n) * KNN + k;
    pOut[base]                    = acc[o][0];
    pOut[base + (long)NPTS * KNN] = acc[o][1];
    pOut[base + 2L * NPTS * KNN]  = acc[o][2];
    const float nrm = sqrtf(acc[o][0]*acc[o][0] + acc[o][1]*acc[o][1] + acc[o][2]*acc[o][2]) + VEPS;
    atomicAdd(&sStat[o * 2 + 0], nrm);
    atomicAdd(&sStat[o * 2 + 1], nrm * nrm);
  }
  __syncthreads();
  if (threadIdx.x < VCOUT * 2) atomicAdd(&stats[threadIdx.x], sStat[threadIdx.x]);
}

// ---------------------------------------------------------------------------
// VN layer pass 2: recompute d = Wd * feat, apply batchnorm (from stats) and
// vector-neuron leaky-relu, in place over the p buffer.
// ---------------------------------------------------------------------------
template<int CIN0, bool GF>
__global__ __launch_bounds__(256) void vn_fin_kernel(
    const float* __restrict__ src, long srcBStride,
    const int* __restrict__ idx,
    const float* __restrict__ Wd,
    float* __restrict__ pInOut,
    const float* __restrict__ stats,
    float invCnt)
{
  constexpr int CIN = GF ? 2 * CIN0 : CIN0;
  __shared__ float sW[VCOUT * CIN];
  __shared__ float sMu[VCOUT];
  __shared__ float sIsd[VCOUT];
  for (int i = threadIdx.x; i < VCOUT * CIN; i += 256) sW[i] = Wd[i];
  if (threadIdx.x < VCOUT) {
    const float mu = stats[threadIdx.x * 2] * invCnt;
    const float var = stats[threadIdx.x * 2 + 1] * invCnt - mu * mu;
    sMu[threadIdx.x] = mu;
    sIsd[threadIdx.x] = rsqrtf(var + BNEPS);
  }
  __syncthreads();

  const long t = (long)blockIdx.x * 256 + threadIdx.x;
  const int b = (int)(t / ((long)NPTS * KNN));
  const int rem = (int)(t % ((long)NPTS * KNN));
  const int n = rem / KNN;
  const int k = rem % KNN;

  float dacc[VCOUT][3];
  #pragma unroll
  for (int o = 0; o < VCOUT; ++o) { dacc[o][0] = 0.f; dacc[o][1] = 0.f; dacc[o][2] = 0.f; }

  if constexpr (GF) {
    const int j = idx[t];
    const float* Xb = src + (long)b * srcBStride;
    for (int c = 0; c < CIN0; ++c) {
      const float* base = Xb + (long)(c * 3) * NPTS;
      const float f0 = base[j]            - base[n];
      const float f1 = base[NPTS + j]     - base[NPTS + n];
      const float f2 = base[2 * NPTS + j] - base[2 * NPTS + n];
      #pragma unroll
      for (int o = 0; o < VCOUT; ++o) {
        const float w = sW[o * CIN + c];
        dacc[o][0] += w * f0; dacc[o][1] += w * f1; dacc[o][2] += w * f2;
      }
    }
    for (int c = 0; c < CIN0; ++c) {
      const float* base = Xb + (long)(c * 3) * NPTS;
      const float f0 = base[n], f1 = base[NPTS + n], f2 = base[2 * NPTS + n];
      #pragma unroll
      for (int o = 0; o < VCOUT; ++o) {
        const float w = sW[o * CIN + CIN0 + c];
        dacc[o][0] += w * f0; dacc[o][1] += w * f1; dacc[o][2] += w * f2;
      }
    }
  } else {
    for (int c = 0; c < CIN0; ++c) {
      const long base = (((long)(b * CIN0 + c) * 3) * NPTS + n) * KNN + k;
      const float f0 = src[base];
      const float f1 = src[base + (long)NPTS * KNN];
      const float f2 = src[base + 2L * NPTS * KNN];
      #pragma unroll
      for (int o = 0; o < VCOUT; ++o) {
        const float w = sW[o * CIN + c];
        dacc[o][0] += w * f0; dacc[o][1] += w * f1; dacc[o][2] += w * f2;
      }
    }
  }

  #pragma unroll
  for (int o = 0; o < VCOUT; ++o) {
    const long base = (((long)(b * VCOUT + o) * 3) * NPTS + n) * KNN + k;
    float p0 = pInOut[base];
    float p1 = pInOut[base + (long)NPTS * KNN];
    float p2 = pInOut[base + 2L * NPTS * KNN];
    const float nrm = sqrtf(p0*p0 + p1*p1 + p2*p2) + VEPS;
    const float sc = (nrm - sMu[o]) * sIsd[o] / nrm;       // batchnorm on norm
    p0 *= sc; p1 *= sc; p2 *= sc;
    const float d0 = dacc[o][0], d1 = dacc[o][1], d2 = dacc[o][2];
    const float dot = p0*d0 + p1*d1 + p2*d2;
    const float dsq = d0*d0 + d1*d1 + d2*d2 + VEPS;
    const float f = (dot < 0.f) ? (dot / dsq) : 0.f;
    pInOut[base]                    = SLOPE*p0 + (1.f-SLOPE)*(p0 - f*d0);
    pInOut[base + (long)NPTS * KNN] = SLOPE*p1 + (1.f-SLOPE)*(p1 - f*d1);
    pInOut[base + 2L * NPTS * KNN]  = SLOPE*p2 + (1.f-SLOPE)*(p2 - f*d2);
  }
}

// ---------------------------------------------------------------------------
// mean over K: h (B,63,N,K) -> out view with per-batch stride (into x123)
// ---------------------------------------------------------------------------
__global__ __launch_bounds__(256) void mean_k_kernel(
    const float* __restrict__ h, float* __restrict__ outp, long outBStride)
{
  const long t = (long)blockIdx.x * 256 + threadIdx.x;   // t < B*63*N
  const int b = (int)(t / ((long)63 * NPTS));
  const int r = (int)(t % ((long)63 * NPTS));
  const float* s = h + ((long)b * 63 * NPTS + r) * KNN;
  float acc = 0.f;
  #pragma unroll
  for (int k = 0; k < KNN; ++k) acc += s[k];
  outp[(long)b * outBStride + r] = acc * (1.f / KNN);
}

// ---------------------------------------------------------------------------
// Layer 6 pass 1: p = Wf6(1024x63) * x123 per (b,d,n) column via WMMA fp32;
// accumulate per-channel norm stats. Recomputed in pass 2 instead of storing
// the 402MB p tensor (GEMM is cheap, HBM traffic is not).
// ---------------------------------------------------------------------------
__global__ __launch_bounds__(256) void l6_stats_kernel(
    const float* __restrict__ X,   // (B,63,3,N)
    const float* __restrict__ Wf,  // (1024,63)
    float* __restrict__ stats)     // (1024,2)
{
  const int b = blockIdx.z;
  const int tid = threadIdx.x;
  const int wave = tid >> 5, lane = tid & 31;
  const int laneHalf = lane & 15;
  const int hi = (lane >= 16) ? 1 : 0;
  const int oBase = (blockIdx.y * 8 + wave) * 16;
  const float* Xb = X + (long)b * 189 * NPTS;

  v2f afrag[16];
  {
    const int orow = oBase + laneHalf;
    #pragma unroll
    for (int c = 0; c < 16; ++c) {
      const int k0 = c * 4 + hi * 2;
      afrag[c].x = (k0     < 63) ? Wf[orow * 63 + k0]     : 0.f;
      afrag[c].y = (k0 + 1 < 63) ? Wf[orow * 63 + k0 + 1] : 0.f;
    }
  }

  float s1[8], s2[8];
  #pragma unroll
  for (int g = 0; g < 8; ++g) { s1[g] = 0.f; s2[g] = 0.f; }

  for (int tIt = 0; tIt < 16; ++tIt) {
    const int n = (blockIdx.x * 16 + tIt) * 16 + laneHalf;
    v8f accD[3];
    #pragma unroll
    for (int dd = 0; dd < 3; ++dd) {
      v8f acc = {0.f,0.f,0.f,0.f,0.f,0.f,0.f,0.f};
      #pragma unroll
      for (int c = 0; c < 16; ++c) {
        const int k0 = c * 4 + hi * 2;
        v2f bfrag;
        bfrag.x = (k0     < 63) ? Xb[((long)(k0    ) * 3 + dd) * NPTS + n] : 0.f;
        bfrag.y = (k0 + 1 < 63) ? Xb[((long)(k0 + 1) * 3 + dd) * NPTS + n] : 0.f;
        acc = wmma_f32_16x16x4(afrag[c], bfrag, acc);
      }
      accD[dd] = acc;
    }
    #pragma unroll
    for (int g = 0; g < 8; ++g) {
      const float nrm = sqrtf(accD[0][g]*accD[0][g] + accD[1][g]*accD[1][g] + accD[2][g]*accD[2][g]) + VEPS;
      s1[g] += nrm; s2[g] += nrm * nrm;
    }
  }

  #pragma unroll
  for (int g = 0; g < 8; ++g) {
    float a = s1[g], c2 = s2[g];
    #pragma unroll
    for (int off = 1; off <= 8; off <<= 1) {
      a  += __shfl_xor(a, off, 32);
      c2 += __shfl_xor(c2, off, 32);
    }
    if (laneHalf == 0) {
      const int o = oBase + g + hi * 8;
      atomicAdd(&stats[o * 2 + 0], a);
      atomicAdd(&stats[o * 2 + 1], c2);
    }
  }
}

// ---------------------------------------------------------------------------
// Layer 6 pass 2: recompute p via WMMA, BN + VN-leaky with broadcast d
// (Wd6 is 1x63), fold mean over N straight into d_out via float atomics.
// ---------------------------------------------------------------------------
__global__ __launch_bounds__(256) void l6_fin_kernel(
    const float* __restrict__ X,
    const float* __restrict__ Wf,
    const float* __restrict__ Wd,     // (63)
    const float* __restrict__ stats,  // (1024,2)
    float* __restrict__ outp)         // (B,1024,3), pre-zeroed
{
  __shared__ float sWd[63];
  __shared__ float sDvec[3][16];
  const int b = blockIdx.z;
  const int tid = threadIdx.x;
  const int wave = tid >> 5, lane = tid & 31;
  const int laneHalf = lane & 15;
  const int hi = (lane >= 16) ? 1 : 0;
  const int oBase = (blockIdx.y * 8 + wave) * 16;
  const float* Xb = X + (long)b * 189 * NPTS;
  if (tid < 63) sWd[tid] = Wd[tid];

  v2f afrag[16];
  {
    const int orow = oBase + laneHalf;
    #pragma unroll
    for (int c = 0; c < 16; ++c) {
      const int k0 = c * 4 + hi * 2;
      afrag[c].x = (k0     < 63) ? Wf[orow * 63 + k0]     : 0.f;
      afrag[c].y = (k0 + 1 < 63) ? Wf[orow * 63 + k0 + 1] : 0.f;
    }
  }

  const float invCnt = 1.f / (float)(BATCH * NPTS);
  float mu[8], isd[8];
  #pragma unroll
  for (int g = 0; g < 8; ++g) {
    const int o = oBase + g + hi * 8;
    const float m = stats[o * 2] * invCnt;
    const float v = stats[o * 2 + 1] * invCnt - m * m;
    mu[g] = m; isd[g] = rsqrtf(v + BNEPS);
  }

  float macc[8][3];
  #pragma unroll
  for (int g = 0; g < 8; ++g) { macc[g][0] = 0.f; macc[g][1] = 0.f; macc[g][2] = 0.f; }

  for (int tIt = 0; tIt < 16; ++tIt) {
    const int nBase = (blockIdx.x * 16 + tIt) * 16;
    __syncthreads();
    if (tid < 48) {                                    // d vector, o-independent
      const int dd = tid / 16, col = tid % 16;
      float s = 0.f;
      for (int c = 0; c < 63; ++c) s += sWd[c] * Xb[((long)c * 3 + dd) * NPTS + nBase + col];
      sDvec[dd][col] = s;
    }
    __syncthreads();

    const int n = nBase + laneHalf;
    v8f accD[3];
    #pragma unroll
    for (int dd = 0; dd < 3; ++dd) {
      v8f acc = {0.f,0.f,0.f,0.f,0.f,0.f,0.f,0.f};
      #pragma unroll
      for (int c = 0; c < 16; ++c) {
        const int k0 = c * 4 + hi * 2;
        v2f bfrag;
        bfrag.x = (k0     < 63) ? Xb[((long)(k0    ) * 3 + dd) * NPTS + n] : 0.f;
        bfrag.y = (k0 + 1 < 63) ? Xb[((long)(k0 + 1) * 3 + dd) * NPTS + n] : 0.f;
        acc = wmma_f32_16x16x4(afrag[c], bfrag, acc);
      }
      accD[dd] = acc;
    }

    const float d0 = sDvec[0][laneHalf], d1 = sDvec[1][laneHalf], d2 = sDvec[2][laneHalf];
    const float dsq = d0*d0 + d1*d1 + d2*d2 + VEPS;
    #pragma unroll
    for (int g = 0; g < 8; ++g) {
      float p0 = accD[0][g], p1 = accD[1][g], p2 = accD[2][g];
      const float nrm = sqrtf(p0*p0 + p1*p1 + p2*p2) + VEPS;
      const float sc = (nrm - mu[g]) * isd[g] / nrm;
      p0 *= sc; p1 *= sc; p2 *= sc;
      const float dot = p0*d0 + p1*d1 + p2*d2;
      const float f = (dot < 0.f) ? (dot / dsq) : 0.f;
      macc[g][0] += SLOPE*p0 + (1.f-SLOPE)*(p0 - f*d0);
      macc[g][1] += SLOPE*p1 + (1.f-SLOPE)*(p1 - f*d1);
      macc[g][2] += SLOPE*p2 + (1.f-SLOPE)*(p2 - f*d2);
    }
  }

  #pragma unroll
  for (int g = 0; g < 8; ++g) {
    float m0 = macc[g][0], m1 = macc[g][1], m2 = macc[g][2];
    #pragma unroll
    for (int off = 1; off <= 8; off <<= 1) {
      m0 += __shfl_xor(m0, off, 32);
      m1 += __shfl_xor(m1, off, 32);
      m2 += __shfl_xor(m2, off, 32);
    }
    if (laneHalf == 0) {
      const int o = oBase + g + hi * 8;
      atomicAdd(&outp[((long)b * FD + o) * 3 + 0], m0 * (1.f / NPTS));
      atomicAdd(&outp[((long)b * FD + o) * 3 + 1], m1 * (1.f / NPTS));
      atomicAdd(&outp[((long)b * FD + o) * 3 + 2], m2 * (1.f / NPTS));
    }
  }
}

// ---------------------------------------------------------------------------
extern "C" void kernel_launch(void* const* d_in, const int* in_sizes, int n_in,
                              void* d_out, int out_size, void* d_ws, size_t ws_size,
                              hipStream_t stream)
{
  (void)in_sizes; (void)n_in; (void)out_size; (void)ws_size;
  const float* x   = (const float*)d_in[0];
  const float* Wf1 = (const float*)d_in[1];
  const float* Wd1 = (const float*)d_in[2];
  const float* Wf2 = (const float*)d_in[3];
  const float* Wd2 = (const float*)d_in[4];
  const float* Wf3 = (const float*)d_in[5];
  const float* Wd3 = (const float*)d_in[6];
  const float* Wf4 = (const float*)d_in[7];
  const float* Wd4 = (const float*)d_in[8];
  const float* Wf5 = (const float*)d_in[9];
  const float* Wd5 = (const float*)d_in[10];
  const float* Wf6 = (const float*)d_in[11];
  const float* Wd6 = (const float*)d_in[12];
  float* outp = (float*)d_out;

  // workspace layout (floats)
  float* ws = (float*)d_ws;
  const long HSZ = (long)BATCH * VCOUT * 3 * NPTS * KNN;   // 41,287,680
  float* h1   = ws;
  float* h2   = h1 + HSZ;
  float* x123 = h2 + HSZ;                                  // (B,63,3,N) = B*189*N
  float* xx   = x123 + (long)BATCH * 189 * NPTS;
  float* statsMid = xx + (long)BATCH * NPTS;               // 5 layers x 64
  float* stats6   = statsMid + 5 * 64;                     // 1024 x 2
  int*   idx  = (int*)(stats6 + 2048);                     // B*N*K ints

  hipMemsetAsync(statsMid, 0, (5 * 64 + 2048) * sizeof(float), stream);
  hipMemsetAsync(outp, 0, (size_t)BATCH * FD * 3 * sizeof(float), stream);

  const size_t knnLds = (size_t)(16 * NPTS + 16) * sizeof(float);   // 256KB strip
  hipFuncSetAttribute(reinterpret_cast<const void*>(&knn_topk_kernel<3>),
                      hipFuncAttributeMaxDynamicSharedMemorySize, (int)knnLds);
  hipFuncSetAttribute(reinterpret_cast<const void*>(&knn_topk_kernel<63>),
                      hipFuncAttributeMaxDynamicSharedMemorySize, (int)knnLds);

  const long Mvn = (long)BATCH * NPTS * KNN;               // 655360 = 2560*256
  const int vnBlocks = (int)(Mvn / 256);
  const float invCntMid = 1.f / (float)Mvn;
  const int sqBlocks = BATCH * NPTS / 256;
  const int meanBlocks = (int)((long)BATCH * 63 * NPTS / 256);
  dim3 knnGrid(NPTS / 16, BATCH);
  dim3 g6(16, 8, BATCH);
  const long bsX  = (long)3 * NPTS;
  const long bsXL = (long)189 * NPTS;

  // ---- stage 1 (C3 = 3) ----
  sqnorm_kernel<<<sqBlocks, 256, 0, stream>>>(x, bsX, 3, xx);
  knn_topk_kernel<3><<<knnGrid, 256, knnLds, stream>>>(x, bsX, xx, idx);
  vn_p_kernel<1, true><<<vnBlocks, 256, 0, stream>>>(x, bsX, idx, Wf1, h1, statsMid + 0 * 64);
  vn_fin_kernel<1, true><<<vnBlocks, 256, 0, stream>>>(x, bsX, idx, Wd1, h1, statsMid + 0 * 64, invCntMid);
  vn_p_kernel<VCOUT, false><<<vnBlocks, 256, 0, stream>>>(h1, 0, idx, Wf2, h2, statsMid + 1 * 64);
  vn_fin_kernel<VCOUT, false><<<vnBlocks, 256, 0, stream>>>(h1, 0, idx, Wd2, h2, statsMid + 1 * 64, invCntMid);
  mean_k_kernel<<<meanBlocks, 256, 0, stream>>>(h2, x123, bsXL);                 // x1 -> channels 0..20

  // ---- stage 2 (C3 = 63) ----
  const float* x1v = x123;
  sqnorm_kernel<<<sqBlocks, 256, 0, stream>>>(x1v, bsXL, 63, xx);
  knn_topk_kernel<63><<<knnGrid, 256, knnLds, stream>>>(x1v, bsXL, xx, idx);
  vn_p_kernel<VCOUT, true><<<vnBlocks, 256, 0, stream>>>(x1v, bsXL, idx, Wf3, h1, statsMid + 2 * 64);
  vn_fin_kernel<VCOUT, true><<<vnBlocks, 256, 0, stream>>>(x1v, bsXL, idx, Wd3, h1, statsMid + 2 * 64, invCntMid);
  vn_p_kernel<VCOUT, false><<<vnBlocks, 256, 0, stream>>>(h1, 0, idx, Wf4, h2, statsMid + 3 * 64);
  vn_fin_kernel<VCOUT, false><<<vnBlocks, 256, 0, stream>>>(h1, 0, idx, Wd4, h2, statsMid + 3 * 64, invCntMid);
  mean_k_kernel<<<meanBlocks, 256, 0, stream>>>(h2, x123 + (long)63 * NPTS, bsXL); // x2 -> channels 21..41

  // ---- stage 3 (C3 = 63) ----
  const float* x2v = x123 + (long)63 * NPTS;
  sqnorm_kernel<<<sqBlocks, 256, 0, stream>>>(x2v, bsXL, 63, xx);
  knn_topk_kernel<63><<<knnGrid, 256, knnLds, stream>>>(x2v, bsXL, xx, idx);
  vn_p_kernel<VCOUT, true><<<vnBlocks, 256, 0, stream>>>(x2v, bsXL, idx, Wf5, h1, statsMid + 4 * 64);
  vn_fin_kernel<VCOUT, true><<<vnBlocks, 256, 0, stream>>>(x2v, bsXL, idx, Wd5, h1, statsMid + 4 * 64, invCntMid);
  mean_k_kernel<<<meanBlocks, 256, 0, stream>>>(h1, x123 + (long)126 * NPTS, bsXL); // x3 -> channels 42..62

  // ---- layer 6: 63 -> 1024 WMMA GEMM, two passes, mean into d_out ----
  l6_stats_kernel<<<g6, 256, 0, stream>>>(x123, Wf6, stats6);
  l6_fin_kernel<<<g6, 256, 0, stream>>>(x123, Wf6, Wd6, stats6, outp);
}